// LabelWiseGraphMLC_26534307955167
// MI455X (gfx1250) — compile-verified
//
#include <hip/hip_runtime.h>

typedef __attribute__((ext_vector_type(2))) float v2f;
typedef __attribute__((ext_vector_type(4))) float v4f;
typedef __attribute__((ext_vector_type(8))) float v8f;

#define B_DIM 32
#define L_DIM 4096
#define D_DIM 1024

// -------------------------------------------------------------------------
// Phase 1: doc_label[row] = dot(doc_rep[row, :], W[row % L, :]) + bias[row % L]
// row in [0, B*L). One wave32 per row, float4 vector loads, shuffle reduce.
// Pure HBM streaming (512 MiB doc_rep) -> this kernel is the runtime floor.
// -------------------------------------------------------------------------
__global__ __launch_bounds__(256) void rowdot_bias_kernel(
    const float* __restrict__ doc, const float* __restrict__ W,
    const float* __restrict__ bias, float* __restrict__ doc_label) {
  const int wave = threadIdx.x >> 5;
  const int lane = threadIdx.x & 31;
  const long long row = (long long)blockIdx.x * 8 + wave;  // 8 waves / block
  const int l = (int)(row & (L_DIM - 1));

  const float* __restrict__ dptr = doc + row * D_DIM;
  const float* __restrict__ wptr = W + (long long)l * D_DIM;

  float acc = 0.0f;
#pragma unroll
  for (int i = 0; i < D_DIM / (32 * 4); ++i) {   // 8 iterations of b128 loads
    v4f d = *(const v4f*)(dptr + i * 128 + lane * 4);
    v4f w = *(const v4f*)(wptr + i * 128 + lane * 4);
    acc = fmaf(d[0], w[0], acc);
    acc = fmaf(d[1], w[1], acc);
    acc = fmaf(d[2], w[2], acc);
    acc = fmaf(d[3], w[3], acc);
  }
  // wave32 reduction (CDNA5 is wave32 — do NOT use width 64)
#pragma unroll
  for (int off = 16; off > 0; off >>= 1) acc += __shfl_xor(acc, off, 32);

  if (lane == 0) doc_label[row] = acc + bias[l];
}

// -------------------------------------------------------------------------
// Phase 2: out = sigmoid(doc_label @ G), M=32, N=4096, K=4096, fp32.
// One wave per 16x16 output tile via V_WMMA_F32_16X16X4_F32 (K-step 4).
// 512 tiles total -> grid 64 blocks x 8 waves.
//
// A frag (16x4 f32, 2 VGPR): lanes 0-15 hold K={k0,k0+1}, lanes 16-31 hold
//   K={k0+2,k0+3}, M = lane%16.
// B frag (4x16 f32, 2 VGPR): same K split, N = lane%16.
// C/D (16x16 f32, 8 VGPR): VGPR r -> row r (lanes 0-15) / r+8 (lanes 16-31).
// -------------------------------------------------------------------------
__global__ __launch_bounds__(256) void gemm_sigmoid_wmma_kernel(
    const float* __restrict__ A,   // [32, 4096] doc_label (workspace)
    const float* __restrict__ G,   // [4096, 4096]
    float* __restrict__ out) {     // [32, 4096]
  const int wave = threadIdx.x >> 5;
  const int lane = threadIdx.x & 31;
  const int tile = blockIdx.x * 8 + wave;  // 0..511
  const int nTile = tile & 255;            // 256 tiles along N
  const int mTile = tile >> 8;             // 2 tiles along M

  const int mn_lane = lane & 15;           // row for A / col for B
  const int khalf = (lane >> 4) * 2;       // 0 for lanes 0-15, 2 for 16-31

  const float* __restrict__ aRow = A + (long long)(mTile * 16 + mn_lane) * L_DIM;
  const float* __restrict__ gCol = G + (long long)(nTile * 16 + mn_lane);

  v8f c = {0.f, 0.f, 0.f, 0.f, 0.f, 0.f, 0.f, 0.f};

#pragma unroll 4
  for (int k0 = 0; k0 < L_DIM; k0 += 4) {
    v2f a, b;
    // A: 64-bit contiguous load per lane
    a[0] = aRow[k0 + khalf + 0];
    a[1] = aRow[k0 + khalf + 1];
    // B: lanes 0-15 read 16 contiguous floats of a G row (coalesced 64B)
    b[0] = gCol[(long long)(k0 + khalf + 0) * L_DIM];
    b[1] = gCol[(long long)(k0 + khalf + 1) * L_DIM];
    // (neg_a, A, neg_b, B, c_mod, C, reuse_a, reuse_b)
    c = __builtin_amdgcn_wmma_f32_16x16x4_f32(false, a, false, b,
                                              (short)0, c, false, false);
  }

  // Fused sigmoid + store. VGPR r -> output row mTile*16 + r + (lane>=16)*8.
  const int colOut = nTile * 16 + mn_lane;
  const int rowBase = mTile * 16 + (lane >> 4) * 8;
#pragma unroll
  for (int r = 0; r < 8; ++r) {
    float x = c[r];
    float s = 1.0f / (1.0f + __expf(-x));
    out[(long long)(rowBase + r) * L_DIM + colOut] = s;
  }
}

// -------------------------------------------------------------------------
// Inputs (setup_inputs order): doc_rep [B,L,D] f32, W [L,D] f32, b [L] f32,
// G [L,L] f32. Output: [B,L] f32. Workspace: doc_label [B,L] f32 = 512 KiB.
// -------------------------------------------------------------------------
extern "C" void kernel_launch(void* const* d_in, const int* in_sizes, int n_in,
                              void* d_out, int out_size, void* d_ws, size_t ws_size,
                              hipStream_t stream) {
  const float* doc = (const float*)d_in[0];
  const float* W   = (const float*)d_in[1];
  const float* b   = (const float*)d_in[2];
  const float* G   = (const float*)d_in[3];
  float* out       = (float*)d_out;
  float* doc_label = (float*)d_ws;  // B*L floats = 512 KiB

  // Phase 1: B*L = 131072 rows, 8 waves (rows) per 256-thread block.
  const int rows = B_DIM * L_DIM;
  rowdot_bias_kernel<<<rows / 8, 256, 0, stream>>>(doc, W, b, doc_label);

  // Phase 2: 512 output tiles of 16x16, 8 waves per block -> 64 blocks.
  gemm_sigmoid_wmma_kernel<<<64, 256, 0, stream>>>(doc_label, G, out);
}